// GNN_17523466567891
// MI455X (gfx1250) — compile-verified
//
#include <hip/hip_runtime.h>
#include <math.h>

typedef float v2f __attribute__((ext_vector_type(2)));
typedef float v8f __attribute__((ext_vector_type(8)));

__device__ __forceinline__ float leaky(float v) { return v > 0.0f ? v : 0.01f * v; }

// ---------------- helpers ----------------
__global__ void k_zero(float* __restrict__ p, long long n) {
  long long i = (long long)blockIdx.x * blockDim.x + threadIdx.x;
  if (i < n) p[i] = 0.0f;
}

// degree count (excluding self loop, added later)
__global__ void k_degree(const long long* __restrict__ dst, float* __restrict__ deg, long long E) {
  long long e = (long long)blockIdx.x * blockDim.x + threadIdx.x;
  if (e < E) atomicAdd(&deg[dst[e]], 1.0f);
}

// dinv = rsqrt(deg+1); repurpose deg buffer as agg1, seeded with self-loop term x[n]/deg
__global__ void k_dinv(const float* __restrict__ x, float* __restrict__ deg_agg,
                       float* __restrict__ dinv, int N) {
  int n = blockIdx.x * blockDim.x + threadIdx.x;
  if (n >= N) return;
  float d = deg_agg[n] + 1.0f;       // +1 self loop
  dinv[n] = rsqrtf(d);
  deg_agg[n] = x[n] / d;             // self-loop contribution to layer-1 aggregation
}

// layer-1 scalar edge scatter: agg1[d] += dinv[s]*dinv[d]*x[s]
__global__ void k_edge1(const long long* __restrict__ src, const long long* __restrict__ dst,
                        const float* __restrict__ x, const float* __restrict__ dinv,
                        float* __restrict__ agg1, long long E) {
  long long e = (long long)blockIdx.x * blockDim.x + threadIdx.x;
  if (e >= E) return;
  long long s = src[e], d = dst[e];
  atomicAdd(&agg1[d], dinv[s] * dinv[d] * x[s]);
}

// h1 = x + leaky(agg1*W1 + b1); seed layer-2 aggregation with self-loop term h1*dinv^2
__global__ void k_h1(const float* __restrict__ x, const float* __restrict__ agg1,
                     const float* __restrict__ dinv, const float* __restrict__ W1,
                     const float* __restrict__ b1, float* __restrict__ h1,
                     float* __restrict__ agg2, long long NC) {
  long long i = (long long)blockIdx.x * blockDim.x + threadIdx.x;
  if (i >= NC) return;
  int c = (int)(i & 31);
  long long n = i >> 5;
  float t = leaky(agg1[n] * W1[c] + b1[c]);
  float v = x[n] + t;
  h1[i] = v;
  float di = dinv[n];
  agg2[i] = v * di * di;
}

// layer-2 edge scatter, one lane per (edge, channel): lanes of a wave share one edge,
// so edge loads are wave-uniform and the 32 h1/agg2 accesses are fully coalesced.
__global__ void k_edge2(const long long* __restrict__ src, const long long* __restrict__ dst,
                        const float* __restrict__ dinv, const float* __restrict__ h1,
                        float* __restrict__ agg2, long long E32) {
  long long t = (long long)blockIdx.x * blockDim.x + threadIdx.x;
  if (t >= E32) return;
  int c = (int)(t & 31);
  long long e = t >> 5;
  long long s = src[e], d = dst[e];
  float w = dinv[s] * dinv[d];
  atomicAdd(&agg2[d * 32 + c], w * h1[s * 32 + c]);
}

// Dense [N,32] @ [32,32] via V_WMMA_F32_16X16X4_F32, fused bias+leaky+residual.
// One wave per 16-node tile; in-place h2 over agg2 (tile is wave-exclusive).
__global__ void k_gemm_wmma(float* h_io /* in: agg2, out: h2 (aliased, no restrict) */,
                            const float* __restrict__ h1, const float* __restrict__ W2,
                            const float* __restrict__ b2, int N) {
  int lane = threadIdx.x & 31;
  int wave = threadIdx.x >> 5;
  int tile = blockIdx.x * 4 + wave;
  int node0 = tile * 16;
  if (node0 >= N) return;            // wave-uniform exit; active waves keep EXEC all-ones

  int row  = lane & 15;              // A row (M) / B column (N) within half
  int half = lane >> 4;              // K sub-pair selector for 32-bit fragments

  v8f acc0 = {};                     // output columns 0..15
  v8f acc1 = {};                     // output columns 16..31

  const float* arow = h_io + (long long)(node0 + row) * 32;

#pragma unroll
  for (int kk = 0; kk < 8; ++kk) {
    int kb = kk * 4 + 2 * half;      // A: V0={K0,K2}, V1={K1,K3} per ISA 32-bit layout
    v2f a;  a.x  = arow[kb];               a.y  = arow[kb + 1];
    v2f b0; b0.x = W2[kb * 32 + row];      b0.y = W2[(kb + 1) * 32 + row];
    v2f b1; b1.x = W2[kb * 32 + 16 + row]; b1.y = W2[(kb + 1) * 32 + 16 + row];
    acc0 = __builtin_amdgcn_wmma_f32_16x16x4_f32(false, a, false, b0, (short)0, acc0, false, false);
    acc1 = __builtin_amdgcn_wmma_f32_16x16x4_f32(false, a, false, b1, (short)0, acc1, false, false);
  }

  // C/D layout: VGPR j, lanes 0-15 -> M=j, lanes 16-31 -> M=j+8, N=lane%16
#pragma unroll
  for (int j = 0; j < 8; ++j) {
    long long node = node0 + j + 8 * half;
    long long base = node * 32;
    {
      int col = row;
      float v = leaky(acc0[j] + b2[col]);
      h_io[base + col] = h1[base + col] + v;
    }
    {
      int col = 16 + row;
      float v = leaky(acc1[j] + b2[col]);
      h_io[base + col] = h1[base + col] + v;
    }
  }
}

// Per-graph mean pool over h2, then 32->1 FC + sigmoid. One block per graph.
__global__ void k_pool_fc(const float* __restrict__ h2, const float* __restrict__ fcW,
                          const float* __restrict__ fcb, float* __restrict__ out, int npg) {
  int g = blockIdx.x;
  int t = threadIdx.x;
  int c = t & 31, grp = t >> 5;      // 8 node-groups x 32 channels
  const float* base = h2 + (long long)g * npg * 32;
  float s = 0.0f;
  for (int n = grp; n < npg; n += 8) s += base[(long long)n * 32 + c];
  __shared__ float sm[256];
  sm[t] = s;
  __syncthreads();
  if (t < 32) {
    float tot = 0.0f;
#pragma unroll
    for (int k = 0; k < 8; ++k) tot += sm[k * 32 + t];
    tot /= (float)npg;
    sm[t] = tot * fcW[t];
  }
  __syncthreads();
  if (t == 0) {
    float acc = fcb[0];
#pragma unroll
    for (int k = 0; k < 32; ++k) acc += sm[k];
    out[g] = 1.0f / (1.0f + expf(-acc));
  }
}

extern "C" void kernel_launch(void* const* d_in, const int* in_sizes, int n_in,
                              void* d_out, int out_size, void* d_ws, size_t ws_size,
                              hipStream_t stream) {
  const float*     x    = (const float*)d_in[0];
  const long long* ei   = (const long long*)d_in[1];   // int64 [2,E]
  // d_in[2] = batch (regular: node/npg), d_in[3] = num_graphs (use out_size)
  const float*     W1   = (const float*)d_in[4];
  const float*     b1   = (const float*)d_in[5];
  const float*     W2   = (const float*)d_in[6];
  const float*     b2   = (const float*)d_in[7];
  const float*     fcW  = (const float*)d_in[8];
  const float*     fcb  = (const float*)d_in[9];
  float*           out  = (float*)d_out;

  const long long N   = in_sizes[0];
  const long long E   = in_sizes[1] / 2;
  const int       G   = out_size;
  const int       npg = (int)(N / G);
  const long long NC  = N * 32;
  const long long E32 = E * 32;

  const long long* src = ei;        // edge_index[0]
  const long long* dst = ei + E;    // edge_index[1]

  // workspace layout (floats): deg/agg1[N] | dinv[N] | h1[32N] | agg2/h2[32N]
  float* wsf  = (float*)d_ws;
  float* agg1 = wsf;                 // also deg
  float* dinv = wsf + N;
  float* h1   = wsf + 2 * N;
  float* agg2 = wsf + 2 * N + NC;    // becomes h2 in place

  const int T = 256;
  dim3 blk(T);

  k_zero<<<dim3((unsigned)((N + T - 1) / T)), blk, 0, stream>>>(agg1, N);
  k_degree<<<dim3((unsigned)((E + T - 1) / T)), blk, 0, stream>>>(dst, agg1, E);
  k_dinv<<<dim3((unsigned)((N + T - 1) / T)), blk, 0, stream>>>(x, agg1, dinv, (int)N);
  k_edge1<<<dim3((unsigned)((E + T - 1) / T)), blk, 0, stream>>>(src, dst, x, dinv, agg1, E);
  k_h1<<<dim3((unsigned)((NC + T - 1) / T)), blk, 0, stream>>>(x, agg1, dinv, W1, b1, h1, agg2, NC);
  k_edge2<<<dim3((unsigned)((E32 + T - 1) / T)), blk, 0, stream>>>(src, dst, dinv, h1, agg2, E32);

  long long tiles = (N + 15) / 16;
  k_gemm_wmma<<<dim3((unsigned)((tiles + 3) / 4)), dim3(128), 0, stream>>>(agg2, h1, W2, b2, (int)N);

  k_pool_fc<<<dim3((unsigned)G), blk, 0, stream>>>(agg2, fcW, fcb, out, npg);
}